// DKTtempoModel_2RNN_58806692217078
// MI455X (gfx1250) — compile-verified
//
#include <hip/hip_runtime.h>
#include <cstdint>

#define B_ 64
#define T_ 512
#define S_ 256
#define E_ 128
#define H_ 128

typedef float v2f __attribute__((ext_vector_type(2)));
typedef float v8f __attribute__((ext_vector_type(8)));

static __device__ __forceinline__ v8f wmma_f32_16x16x4(v2f a, v2f b, v8f c) {
  // D = A(16x4) * B(4x16) + C(16x16), full fp32 (matches reference numerics)
  return __builtin_amdgcn_wmma_f32_16x16x4_f32(false, a, false, b, (short)0, c,
                                               false, false);
}

static __device__ __forceinline__ float sigf(float x) {
  return 1.0f / (1.0f + __expf(-x));
}

// ---------------------------------------------------------------------------
// Kernel 1: per-(b,t) row reductions: mask_d, mask_x, rowsum(x)
// ---------------------------------------------------------------------------
__global__ __launch_bounds__(256) void dkt_maskrow_kernel(
    const float* __restrict__ x, const float* __restrict__ neg_sk,
    const float* __restrict__ delta, int* __restrict__ mask_x,
    int* __restrict__ mask_d, float* __restrict__ rowsum) {
  const int row = blockIdx.x;         // (b*T + t)
  const int tid = threadIdx.x;        // 256 threads
  const float* xr = x + (size_t)row * (2 * S_);
  const float x0 = xr[tid];
  const float x1 = xr[tid + 256];
  const float ns = neg_sk[(size_t)row * S_ + tid];
  const float dl = delta[(size_t)row * S_ + tid];

  __shared__ float ssum[256];
  __shared__ int sfx[256];
  __shared__ int sfd[256];
  ssum[tid] = x0 + x1;
  sfx[tid] = (x0 != 0.0f) || (x1 != 0.0f);
  sfd[tid] = (ns != 0.0f) || (dl != 0.0f);
  __syncthreads();
  for (int s = 128; s > 0; s >>= 1) {
    if (tid < s) {
      ssum[tid] += ssum[tid + s];
      sfx[tid] |= sfx[tid + s];
      sfd[tid] |= sfd[tid + s];
    }
    __syncthreads();
  }
  if (tid == 0) {
    rowsum[row] = ssum[0];
    mask_x[row] = sfx[0];
    mask_d[row] = sfd[0];
  }
}

// ---------------------------------------------------------------------------
// Kernel 2: tempo scan  h = neg_sk*h + delta (masked), parallel over (b,s)
// ---------------------------------------------------------------------------
__global__ __launch_bounds__(256) void dkt_tempo_scan_kernel(
    const float* __restrict__ neg_sk, const float* __restrict__ delta,
    const float* __restrict__ init_tempo, const int* __restrict__ mask_d,
    float* __restrict__ tempo) {
  const int idx = blockIdx.x * blockDim.x + threadIdx.x;  // 0..B*S-1
  const int b = idx >> 8;
  const int s = idx & (S_ - 1);
  float h = init_tempo[idx];
  const int* md = mask_d + b * T_;
  const size_t base = (size_t)b * T_ * S_ + s;
  for (int t = 0; t < T_; ++t) {
    const size_t o = base + (size_t)t * S_;
    if (md[t]) h = neg_sk[o] * h + delta[o];
    tempo[o] = h;
  }
}

// ---------------------------------------------------------------------------
// Kernel 3: count scan  h = rowsum(x_t)*h + x_t (masked), parallel (b,2s)
// ---------------------------------------------------------------------------
__global__ __launch_bounds__(256) void dkt_count_scan_kernel(
    const float* __restrict__ x, const float* __restrict__ rowsum,
    const int* __restrict__ mask_x, float* __restrict__ count) {
  const int idx = blockIdx.x * blockDim.x + threadIdx.x;  // 0..B*2S-1
  const int b = idx >> 9;
  const int j = idx & (2 * S_ - 1);
  float h = 0.0f;
  const int* mx = mask_x + b * T_;
  const float* rs = rowsum + b * T_;
  const size_t base = (size_t)b * T_ * (2 * S_) + j;
  for (int t = 0; t < T_; ++t) {
    const size_t o = base + (size_t)t * (2 * S_);
    if (mx[t]) h = rs[t] * h + x[o];
    count[o] = h;
  }
}

// ---------------------------------------------------------------------------
// WMMA f32 GEMM: C(MxN) = epi(A(MxK) @ W(KxN) + bias)
// 256 threads / 8 waves per block; each wave -> one 16x64 strip.
// EPI 0: identity   EPI 1: exp(-sigmoid(v))   EPI 2: v * aux[row,col]
// ---------------------------------------------------------------------------
template <int EPI>
__global__ __launch_bounds__(256) void dkt_gemm_wmma_kernel(
    const float* __restrict__ A, const float* __restrict__ W,
    const float* __restrict__ bias, float* __restrict__ C,
    const float* __restrict__ aux, int M, int N, int K) {
  const int wave = threadIdx.x >> 5;
  const int lane = threadIdx.x & 31;
  const int half = lane >> 4;
  const int l = lane & 15;
  const int nTiles = N >> 6;
  const int strip = blockIdx.x * 8 + wave;
  const int m0 = (strip / nTiles) << 4;
  const int n0 = (strip % nTiles) << 6;

  v8f acc[4];
#pragma unroll
  for (int j = 0; j < 4; ++j) acc[j] = (v8f){0, 0, 0, 0, 0, 0, 0, 0};

  // A fragment (16x4 f32): lane l holds row m0+l; VGPR0=K+2*half, VGPR1=+1
  const float* arow = A + (size_t)(m0 + l) * K + 2 * half;
  // B fragment (4x16 f32): lane l holds col; VGPR0 = row k+2*half, VGPR1 = +1
  const float* wb = W + (size_t)(2 * half) * N + n0 + l;

  for (int k = 0; k < K; k += 4) {
    v2f a;
    a.x = arow[k];
    a.y = arow[k + 1];
    const float* wk = wb + (size_t)k * N;
#pragma unroll
    for (int j = 0; j < 4; ++j) {
      v2f b;
      b.x = wk[j * 16];
      b.y = wk[j * 16 + N];
      acc[j] = wmma_f32_16x16x4(a, b, acc[j]);
    }
  }

#pragma unroll
  for (int j = 0; j < 4; ++j) {
    const int col = n0 + j * 16 + l;
    const float bv = bias[col];
#pragma unroll
    for (int i = 0; i < 8; ++i) {
      const int row = m0 + i + 8 * half;  // C/D layout: rows i and i+8
      float v = acc[j][i] + bv;
      if (EPI == 1) v = __expf(-sigf(v));
      if (EPI == 2) v = v * aux[(size_t)row * N + col];
      C[(size_t)row * N + col] = v;
    }
  }
}

// ---------------------------------------------------------------------------
// Kernel: pack xc = [embed_x (128) | dotted (256) | d (256)]
// ---------------------------------------------------------------------------
__global__ __launch_bounds__(256) void dkt_pack_xc_kernel(
    const float* __restrict__ ex, const float* __restrict__ dotted,
    const float* __restrict__ dbuf, float* __restrict__ xc) {
  const int idx = blockIdx.x * blockDim.x + threadIdx.x;  // < 32768*640
  const int row = idx / 640;
  const int j = idx - row * 640;
  float v;
  if (j < E_)
    v = ex[(size_t)row * E_ + j];
  else if (j < E_ + S_)
    v = dotted[(size_t)row * S_ + (j - E_)];
  else
    v = dbuf[(size_t)row * S_ + (j - E_ - S_)];
  xc[idx] = v;
}

// ---------------------------------------------------------------------------
// LSTM recurrence: one workgroup, 32 waves.  h in LDS (32KB), c in VGPRs.
// Wave w owns rows m0..m0+15 and h-columns hh0..hh0+15; it computes the
// i/f/g/o gate z-columns for those cells itself (4 accumulators), so the
// gate nonlinearity happens entirely in-register.  z = zpre (precomputed
// xc@W_lstm + b_lstm) + h@U_lstm via f32 WMMA over K=128.
// ---------------------------------------------------------------------------
__global__ __launch_bounds__(1024) void dkt_lstm_kernel(
    const float* __restrict__ zpre, const float* __restrict__ U,
    const int* __restrict__ mask_x, float* __restrict__ hseq) {
  __shared__ float sh_h[B_ * H_];
  const int tid = threadIdx.x;
  const int wave = tid >> 5;
  const int lane = tid & 31;
  const int half = lane >> 4;
  const int l = lane & 15;
  const int m0 = (wave >> 3) << 4;  // 0,16,32,48
  const int hh0 = (wave & 7) << 4;  // 0..112

  for (int i = tid; i < B_ * H_; i += 1024) sh_h[i] = 0.0f;
  float creg[8];
#pragma unroll
  for (int i = 0; i < 8; ++i) creg[i] = 0.0f;
  __syncthreads();

  const float* a_base = sh_h + (m0 + l) * H_ + 2 * half;
  const int hh = hh0 + l;

  for (int t = 0; t < T_; ++t) {
    v8f zi = (v8f){0, 0, 0, 0, 0, 0, 0, 0};
    v8f zf = (v8f){0, 0, 0, 0, 0, 0, 0, 0};
    v8f zg = (v8f){0, 0, 0, 0, 0, 0, 0, 0};
    v8f zo = (v8f){0, 0, 0, 0, 0, 0, 0, 0};
    for (int k = 0; k < H_; k += 4) {
      v2f a;
      a.x = a_base[k];
      a.y = a_base[k + 1];
      const float* u0 = U + (size_t)(k + 2 * half) * (4 * H_) + hh;
      v2f b;
      b.x = u0[0 * H_];            b.y = u0[0 * H_ + 4 * H_];
      zi = wmma_f32_16x16x4(a, b, zi);
      b.x = u0[1 * H_];            b.y = u0[1 * H_ + 4 * H_];
      zf = wmma_f32_16x16x4(a, b, zf);
      b.x = u0[2 * H_];            b.y = u0[2 * H_ + 4 * H_];
      zg = wmma_f32_16x16x4(a, b, zg);
      b.x = u0[3 * H_];            b.y = u0[3 * H_ + 4 * H_];
      zo = wmma_f32_16x16x4(a, b, zo);
    }
    __syncthreads();  // all waves done reading sh_h for this step

#pragma unroll
    for (int i = 0; i < 8; ++i) {
      const int m = m0 + i + 8 * half;
      const float* zr = zpre + ((size_t)m * T_ + t) * (4 * H_);
      const float vi = zi[i] + zr[hh];
      const float vf = zf[i] + zr[H_ + hh];
      const float vg = zg[i] + zr[2 * H_ + hh];
      const float vo = zo[i] + zr[3 * H_ + hh];
      const float cn = sigf(vf) * creg[i] + sigf(vi) * tanhf(vg);
      const float hn = sigf(vo) * tanhf(cn);
      const bool msk = mask_x[m * T_ + t] != 0;
      float hv;
      if (msk) {
        creg[i] = cn;
        hv = hn;
      } else {
        hv = sh_h[m * H_ + hh];  // this lane exclusively owns this cell
      }
      sh_h[m * H_ + hh] = hv;
      hseq[((size_t)m * T_ + t) * H_ + hh] = hv;
    }
    __syncthreads();  // sh_h updated before next step's GEMM
  }
}

// ---------------------------------------------------------------------------
// Final: out[r] = sum_s sigmoid(yln[r,s]) * q[r,s]   (one wave per row)
// ---------------------------------------------------------------------------
__global__ __launch_bounds__(256) void dkt_reduce_out_kernel(
    const float* __restrict__ yln, const float* __restrict__ q,
    float* __restrict__ out) {
  const int wave = threadIdx.x >> 5;
  const int lane = threadIdx.x & 31;
  const int r = blockIdx.x * 8 + wave;
  const float* yr = yln + (size_t)r * S_;
  const float* qr = q + (size_t)r * S_;
  float v = 0.0f;
#pragma unroll
  for (int s = 0; s < S_; s += 32) v += sigf(yr[s + lane]) * qr[s + lane];
  for (int off = 16; off > 0; off >>= 1) v += __shfl_xor(v, off, 32);
  if (lane == 0) out[r] = v;
}

// ---------------------------------------------------------------------------
extern "C" void kernel_launch(void* const* d_in, const int* in_sizes, int n_in,
                              void* d_out, int out_size, void* d_ws,
                              size_t ws_size, hipStream_t stream) {
  const float* x          = (const float*)d_in[0];
  const float* neg_sk     = (const float*)d_in[1];
  const float* delta      = (const float*)d_in[2];
  const float* q          = (const float*)d_in[3];
  const float* init_tempo = (const float*)d_in[4];
  const float* W_f1       = (const float*)d_in[5];
  const float* b_f1       = (const float*)d_in[6];
  const float* W_f2       = (const float*)d_in[7];
  const float* b_f2       = (const float*)d_in[8];
  const float* W_ce       = (const float*)d_in[9];
  const float* b_ce       = (const float*)d_in[10];
  const float* W_xe       = (const float*)d_in[11];
  const float* b_xe       = (const float*)d_in[12];
  const float* W_lstm     = (const float*)d_in[13];
  const float* U_lstm     = (const float*)d_in[14];
  const float* b_lstm     = (const float*)d_in[15];
  const float* W_out      = (const float*)d_in[16];
  const float* b_out      = (const float*)d_in[17];
  float* out = (float*)d_out;

  const int M = B_ * T_;  // 32768
  // Workspace layout (floats), with lifetime-based reuse
  float* ws = (float*)d_ws;
  int* mask_d = (int*)ws;                 // 32768
  int* mask_x = (int*)(ws + M);           // 32768
  float* rowsum = ws + 2 * M;             // 32768
  float* regA = ws + 3 * M;               //  8,388,608: tempo -> dotted
  float* regB = regA + (size_t)M * S_;    //  8,388,608: E -> embed_x
  float* regC = regB + (size_t)M * S_;    // 16,777,216: count -> zpre
  float* regD = regC + (size_t)M * 2 * S_;//  8,388,608: d -> yln
  float* regF = regD + (size_t)M * S_;    // 20,971,520: xc -> hseq

  float* tempo = regA;
  float* Ebuf  = regB;
  float* count = regC;
  float* dbuf  = regD;
  float* dotted = regA;
  float* exbuf = regB;
  float* xc    = regF;
  float* zpre  = regC;
  float* hseq  = regF;
  float* yln   = regD;

  // 1) row masks + row sums
  dkt_maskrow_kernel<<<M, 256, 0, stream>>>(x, neg_sk, delta, mask_x, mask_d,
                                            rowsum);
  // 2) tempo scan, 3) count scan
  dkt_tempo_scan_kernel<<<(B_ * S_) / 256, 256, 0, stream>>>(
      neg_sk, delta, init_tempo, mask_d, tempo);
  dkt_count_scan_kernel<<<(B_ * 2 * S_) / 256, 256, 0, stream>>>(
      x, rowsum, mask_x, count);

  auto gblk = [&](int N) { return (M / 16) * (N / 64) / 8; };

  // 4) E = exp(-sigmoid(tempo @ W_f1 + b_f1))
  dkt_gemm_wmma_kernel<1><<<gblk(S_), 256, 0, stream>>>(
      tempo, W_f1, b_f1, Ebuf, nullptr, M, S_, S_);
  // 5) d = E @ W_f2 + b_f2
  dkt_gemm_wmma_kernel<0><<<gblk(S_), 256, 0, stream>>>(
      Ebuf, W_f2, b_f2, dbuf, nullptr, M, S_, S_);
  // 6) dotted = d * (count @ W_ce + b_ce)
  dkt_gemm_wmma_kernel<2><<<gblk(S_), 256, 0, stream>>>(
      count, W_ce, b_ce, dotted, dbuf, M, S_, 2 * S_);
  // 7) embed_x = x @ W_xe + b_xe
  dkt_gemm_wmma_kernel<0><<<gblk(E_), 256, 0, stream>>>(
      x, W_xe, b_xe, exbuf, nullptr, M, E_, 2 * S_);
  // 8) xc = [embed_x | dotted | d]
  dkt_pack_xc_kernel<<<(M * 640) / 256, 256, 0, stream>>>(exbuf, dotted, dbuf,
                                                          xc);
  // 9) zpre = xc @ W_lstm + b_lstm   (K=640, N=512)
  dkt_gemm_wmma_kernel<0><<<gblk(4 * H_), 256, 0, stream>>>(
      xc, W_lstm, b_lstm, zpre, nullptr, M, 4 * H_, E_ + 2 * S_);
  // 10) serial LSTM recurrence (single WGP)
  dkt_lstm_kernel<<<1, 1024, 0, stream>>>(zpre, U_lstm, mask_x, hseq);
  // 11) yln = hseq @ W_out + b_out
  dkt_gemm_wmma_kernel<0><<<gblk(S_), 256, 0, stream>>>(
      hseq, W_out, b_out, yln, nullptr, M, S_, H_);
  // 12) out = sum(sigmoid(yln) * q)
  dkt_reduce_out_kernel<<<M / 8, 256, 0, stream>>>(yln, q, out);
}